// HeteroConvLayer_51058571214897
// MI455X (gfx1250) — compile-verified
//
#include <hip/hip_runtime.h>

typedef __attribute__((ext_vector_type(2))) float v2f;
typedef __attribute__((ext_vector_type(8))) float v8f;

// address-space(1) = global; forces global_load_b64 instead of flat_load_b64
typedef __attribute__((address_space(1))) const float gfloat;
typedef __attribute__((address_space(1))) const v2f   gv2f;

__device__ __forceinline__ v2f load_global_v2f(const float* p)
{
    return *(const gv2f*)(const gfloat*)p;
}

#define N_WORD  50000
#define N_TOPIC 2000
#define N_DOC   20000
#define D_OUT   128
#define D_WORD  300
#define D_TOPIC 128

// Up to 3 weight sets applied to the same source features in one pass.
struct GemmMulti {
    const float* W[3];
    const float* bias[3];
    float*       out[3];
};

// ---------------------------------------------------------------------------
// Fused multi-weight GEMM: for w in [0,NW): out_w = feat @ W_w^T + b_w
// One block = 256 threads = 8 waves. Block covers 16 rows; wave v covers
// columns [16v, 16v+16). Each wave holds NW 16x16 f32 accumulators.
//
// K-loop is chunked by KC k-steps (4*KC floats): all KC*(1+NW) fragment
// loads for a chunk are issued before any of its KC*NW WMMAs, so the
// scheduler clauses the loads and staggers s_wait_loadcnt, overlapping
// v_wmma_f32_16x16x4_f32 with in-flight loads. Requires d_in % (4*KC) == 0.
// All fragment loads are forced to the GLOBAL path (no FLAT/DScnt coupling).
//
// fp32 WMMA fragment layouts (ISA 7.12.2):
//  A (16x4):  lane = M%16; VGPR0 K = 2*(lane/16), VGPR1 K+1
//  B (4x16):  lane = N%16; VGPR0 K = 2*(lane/16), VGPR1 K+1
//  C (16x16): VGPR i: lanes 0-15 -> M=i, N=lane; lanes 16-31 -> M=i+8
// ---------------------------------------------------------------------------
template <int NW, int KC>
__global__ __launch_bounds__(256) void gemm_wmma_multi(
    const float* __restrict__ feat, int n_src, int d_in, GemmMulti args)
{
    const int lane    = threadIdx.x & 31;
    const int wave    = threadIdx.x >> 5;
    const int rowbase = blockIdx.x << 4;
    const int colbase = wave << 4;          // 8 waves * 16 = 128 = D_OUT
    const int half    = lane >> 4;          // 0: lanes 0-15, 1: lanes 16-31
    const int l16     = lane & 15;

    int arow = rowbase + l16;
    if (arow >= n_src) arow = n_src - 1;    // clamp: keep EXEC all-ones for WMMA
    const float* ap = feat + (size_t)arow * d_in + (half << 1);

    const float* bp[NW];
#pragma unroll
    for (int w = 0; w < NW; ++w)
        bp[w] = args.W[w] + (size_t)(colbase + l16) * d_in + (half << 1);

    v8f acc[NW];
#pragma unroll
    for (int w = 0; w < NW; ++w) acc[w] = (v8f){};

    const int nchunk = d_in / (4 * KC);
    for (int c = 0; c < nchunk; ++c) {
        // ---- load phase: all fragments for this chunk (immediate offsets) ----
        v2f a[KC];
#pragma unroll
        for (int j = 0; j < KC; ++j)
            a[j] = load_global_v2f(ap + 4 * j);

        v2f b[NW][KC];
#pragma unroll
        for (int w = 0; w < NW; ++w)
#pragma unroll
            for (int j = 0; j < KC; ++j)
                b[w][j] = load_global_v2f(bp[w] + 4 * j);

        // ---- compute phase: KC*NW WMMAs over the staged fragments ----
#pragma unroll
        for (int j = 0; j < KC; ++j)
#pragma unroll
            for (int w = 0; w < NW; ++w)
                acc[w] = __builtin_amdgcn_wmma_f32_16x16x4_f32(
                    /*neg_a=*/false, a[j], /*neg_b=*/false, b[w][j],
                    /*c_mod=*/(short)0, acc[w],
                    /*reuse_a=*/false, /*reuse_b=*/false);

        ap += 4 * KC;
#pragma unroll
        for (int w = 0; w < NW; ++w) bp[w] += 4 * KC;
    }

#pragma unroll
    for (int w = 0; w < NW; ++w) {
        const float bv = args.bias[w][colbase + l16];
#pragma unroll
        for (int i = 0; i < 8; ++i) {
            const int m = rowbase + i + (half << 3);
            if (m < n_src)
                args.out[w][(size_t)m * D_OUT + colbase + l16] = acc[w][i] + bv;
        }
    }
}

// ---------------------------------------------------------------------------
// In-degree counts per edge type: cnt[dst[e]] += 1
// ---------------------------------------------------------------------------
__global__ __launch_bounds__(256) void count_deg(
    const int* __restrict__ dst, int ne, float* __restrict__ cnt)
{
    int i = blockIdx.x * blockDim.x + threadIdx.x;
    const int stride = gridDim.x * blockDim.x;
    for (; i < ne; i += stride)
        atomicAdd(&cnt[dst[i]], 1.0f);
}

// ---------------------------------------------------------------------------
// Scatter-mean: out[dst[e]] += Wh[src[e]] * (ew[e] / max(cnt[dst[e]],1))
// One wave per edge; each lane handles 4 of 128 channels (float4 gather,
// 4 fp32 global atomics). 512B coalesced per wave, L2-resident RMW.
// ---------------------------------------------------------------------------
__global__ __launch_bounds__(256) void scatter_mean(
    const float* __restrict__ Wh,
    const int* __restrict__ src, const int* __restrict__ dst,
    const float* __restrict__ ew, const float* __restrict__ cnt,
    int ne, float* __restrict__ out)
{
    const int lane  = threadIdx.x & 31;
    const int wave  = threadIdx.x >> 5;
    const int wstep = (blockDim.x >> 5) * gridDim.x;

    for (int e = blockIdx.x * (blockDim.x >> 5) + wave; e < ne; e += wstep) {
        const int   s     = src[e];
        const int   d     = dst[e];
        const float scale = ew[e] / fmaxf(cnt[d], 1.0f);

        const float4 v = ((const float4*)(Wh + (size_t)s * D_OUT))[lane];
        float* op = out + (size_t)d * D_OUT + (lane << 2);
        atomicAdd(op + 0, v.x * scale);
        atomicAdd(op + 1, v.y * scale);
        atomicAdd(op + 2, v.z * scale);
        atomicAdd(op + 3, v.w * scale);
    }
}

// ---------------------------------------------------------------------------
extern "C" void kernel_launch(void* const* d_in, const int* in_sizes, int n_in,
                              void* d_out, int out_size, void* d_ws, size_t ws_size,
                              hipStream_t stream)
{
    // setup_inputs() dict order
    const float* feat_word  = (const float*)d_in[0];
    const float* feat_topic = (const float*)d_in[1];
    const int*   src_ww = (const int*)d_in[2];
    const int*   dst_ww = (const int*)d_in[3];
    const float* ew_ww  = (const float*)d_in[4];
    const int*   src_wt = (const int*)d_in[5];
    const int*   dst_wt = (const int*)d_in[6];
    const float* ew_wt  = (const float*)d_in[7];
    const int*   src_wd = (const int*)d_in[8];
    const int*   dst_wd = (const int*)d_in[9];
    const float* ew_wd  = (const float*)d_in[10];
    const int*   src_td = (const int*)d_in[11];
    const int*   dst_td = (const int*)d_in[12];
    const float* ew_td  = (const float*)d_in[13];
    const int*   src_tt = (const int*)d_in[14];
    const int*   dst_tt = (const int*)d_in[15];
    const float* ew_tt  = (const float*)d_in[16];
    const float* W_ww = (const float*)d_in[17]; const float* b_ww = (const float*)d_in[18];
    const float* W_wt = (const float*)d_in[19]; const float* b_wt = (const float*)d_in[20];
    const float* W_wd = (const float*)d_in[21]; const float* b_wd = (const float*)d_in[22];
    const float* W_td = (const float*)d_in[23]; const float* b_td = (const float*)d_in[24];
    const float* W_tt = (const float*)d_in[25]; const float* b_tt = (const float*)d_in[26];

    const int ne_ww = in_sizes[2];
    const int ne_wt = in_sizes[5];
    const int ne_wd = in_sizes[8];
    const int ne_td = in_sizes[11];
    const int ne_tt = in_sizes[14];

    // workspace layout (floats)
    float* ws = (float*)d_ws;
    size_t o = 0;
    float* Wh_ww = ws + o; o += (size_t)N_WORD  * D_OUT;
    float* Wh_wt = ws + o; o += (size_t)N_WORD  * D_OUT;
    float* Wh_wd = ws + o; o += (size_t)N_WORD  * D_OUT;
    float* Wh_td = ws + o; o += (size_t)N_TOPIC * D_OUT;
    float* Wh_tt = ws + o; o += (size_t)N_TOPIC * D_OUT;
    float* cnt_ww = ws + o; o += N_WORD;
    float* cnt_wt = ws + o; o += N_TOPIC;
    float* cnt_tt = ws + o; o += N_TOPIC;
    float* cnt_wd = ws + o; o += N_DOC;
    float* cnt_td = ws + o; o += N_DOC;
    const size_t cnt_floats = (size_t)N_WORD + 2 * N_TOPIC + 2 * N_DOC;

    float* out       = (float*)d_out;
    float* out_word  = out;
    float* out_topic = out + (size_t)N_WORD * D_OUT;
    float* out_doc   = out + (size_t)(N_WORD + N_TOPIC) * D_OUT;

    // zero accumulators (graph-capture-safe)
    hipMemsetAsync(d_out, 0, (size_t)out_size * sizeof(float), stream);
    hipMemsetAsync(cnt_ww, 0, cnt_floats * sizeof(float), stream);

    // Phase 1: fused WMMA GEMMs — feat_word read once for 3 weight sets,
    // feat_topic once for 2. KC chosen so d_in % (4*KC) == 0.
    GemmMulti gw_args;
    gw_args.W[0] = W_ww; gw_args.bias[0] = b_ww; gw_args.out[0] = Wh_ww;
    gw_args.W[1] = W_wt; gw_args.bias[1] = b_wt; gw_args.out[1] = Wh_wt;
    gw_args.W[2] = W_wd; gw_args.bias[2] = b_wd; gw_args.out[2] = Wh_wd;

    GemmMulti gt_args;
    gt_args.W[0] = W_td; gt_args.bias[0] = b_td; gt_args.out[0] = Wh_td;
    gt_args.W[1] = W_tt; gt_args.bias[1] = b_tt; gt_args.out[1] = Wh_tt;
    gt_args.W[2] = W_tt; gt_args.bias[2] = b_tt; gt_args.out[2] = Wh_tt;  // unused

    const int gw = (N_WORD  + 15) / 16;
    const int gt = (N_TOPIC + 15) / 16;
    gemm_wmma_multi<3, 5><<<gw, 256, 0, stream>>>(feat_word,  N_WORD,  D_WORD,  gw_args);
    gemm_wmma_multi<2, 4><<<gt, 256, 0, stream>>>(feat_topic, N_TOPIC, D_TOPIC, gt_args);

    // Phase 2: in-degree counts per edge type
    count_deg<<<(ne_ww + 255) / 256, 256, 0, stream>>>(dst_ww, ne_ww, cnt_ww);
    count_deg<<<(ne_wt + 255) / 256, 256, 0, stream>>>(dst_wt, ne_wt, cnt_wt);
    count_deg<<<(ne_wd + 255) / 256, 256, 0, stream>>>(dst_wd, ne_wd, cnt_wd);
    count_deg<<<(ne_td + 255) / 256, 256, 0, stream>>>(dst_td, ne_td, cnt_td);
    count_deg<<<(ne_tt + 255) / 256, 256, 0, stream>>>(dst_tt, ne_tt, cnt_tt);

    // Phase 3: scatter-mean directly into concatenated output
    scatter_mean<<<(ne_ww + 7) / 8, 256, 0, stream>>>(Wh_ww, src_ww, dst_ww, ew_ww, cnt_ww, ne_ww, out_word);
    scatter_mean<<<(ne_wt + 7) / 8, 256, 0, stream>>>(Wh_wt, src_wt, dst_wt, ew_wt, cnt_wt, ne_wt, out_topic);
    scatter_mean<<<(ne_tt + 7) / 8, 256, 0, stream>>>(Wh_tt, src_tt, dst_tt, ew_tt, cnt_tt, ne_tt, out_topic);
    scatter_mean<<<(ne_wd + 7) / 8, 256, 0, stream>>>(Wh_wd, src_wd, dst_wd, ew_wd, cnt_wd, ne_wd, out_doc);
    scatter_mean<<<(ne_td + 7) / 8, 256, 0, stream>>>(Wh_td, src_td, dst_td, ew_td, cnt_td, ne_td, out_doc);
}